// Generator3DLUT_identity_20744692039900
// MI455X (gfx1250) — compile-verified
//
#include <hip/hip_runtime.h>

// 3D LUT trilinear interpolation, MI455X (gfx1250, wave32).
//
// Roofline: 201 MB streamed @ 23.3 TB/s => ~8.6 us floor; ~0.6 GFLOP total.
// Memory-bound gather kernel: WMMA is inapplicable (no dense GEMM structure).
// Strategy: float4-packed LUT (one b128 gather per corner, resident in 192MB L2),
// 4 pixels/lane with b128 vector loads/stores, non-temporal hints on the
// streamed tensors so they don't evict the LUT from cache.

typedef float v4f __attribute__((ext_vector_type(4)));

#define LUT_DIM   33
#define LUT_DIM2  (LUT_DIM * LUT_DIM)
#define LUT_N     (LUT_DIM * LUT_DIM * LUT_DIM)   // 35937 entries
#define HW        (512 * 512)                     // pixels per plane
#define QPI       (HW / 4)                        // quads per image = 65536
#define QPI_SHIFT 16
#define QPI_MASK  (QPI - 1)

// ---- Pre-pass: repack LUT [3][33^3] channel-planar -> [33^3] float4 (RGBx) ----
__global__ __launch_bounds__(256) void lut_pack_kernel(
    const float* __restrict__ lut, v4f* __restrict__ lut4)
{
    int i = blockIdx.x * blockDim.x + threadIdx.x;
    if (i < LUT_N) {
        v4f v;
        v.x = lut[i];                 // R channel
        v.y = lut[i + LUT_N];         // G channel
        v.z = lut[i + 2 * LUT_N];     // B channel
        v.w = 0.0f;
        lut4[i] = v;
    }
}

// ---- Main pass: 4 consecutive pixels per lane, packed-LUT gathers ----
__global__ __launch_bounds__(256) void lut_apply_kernel(
    const float* __restrict__ x, const v4f* __restrict__ lut4,
    float* __restrict__ out, int total_quads)
{
    int tid = blockIdx.x * blockDim.x + threadIdx.x;
    if (tid >= total_quads) return;

    int img = tid >> QPI_SHIFT;          // image index
    int q   = tid & QPI_MASK;            // quad within plane
    size_t base = (size_t)img * 3 * HW + ((size_t)q << 2);

    // Streamed reads: non-temporal (read-once) -> keep L2 for the LUT.
    const v4f r4 = __builtin_nontemporal_load((const v4f*)(x + base));
    const v4f g4 = __builtin_nontemporal_load((const v4f*)(x + base + HW));
    const v4f b4 = __builtin_nontemporal_load((const v4f*)(x + base + 2 * HW));

    v4f outr, outg, outb;
    const float scale = (float)(LUT_DIM - 1);

    #pragma unroll
    for (int j = 0; j < 4; ++j) {
        float rs = r4[j] * scale;
        float gs = g4[j] * scale;
        float bs = b4[j] * scale;

        int r_id = (int)floorf(rs);
        int g_id = (int)floorf(gs);
        int b_id = (int)floorf(bs);
        r_id = min(max(r_id, 0), LUT_DIM - 2);
        g_id = min(max(g_id, 0), LUT_DIM - 2);
        b_id = min(max(b_id, 0), LUT_DIM - 2);

        float rd = rs - (float)r_id;
        float gd = gs - (float)g_id;
        float bd = bs - (float)b_id;
        float rw0 = 1.0f - rd, rw1 = rd;
        float gw0 = 1.0f - gd, gw1 = gd;
        float bw0 = 1.0f - bd, bw1 = bd;

        // LUT flat index: b slowest, g middle, r fastest (matches reference
        // LUT[:, b_id+db, g_id+dg, r_id+dr]).
        int i000 = (b_id * LUT_DIM + g_id) * LUT_DIM + r_id;

        // 8 corners, one b128 gather each (L2-resident).
        v4f c000 = lut4[i000];
        v4f c001 = lut4[i000 + 1];
        v4f c010 = lut4[i000 + LUT_DIM];
        v4f c011 = lut4[i000 + LUT_DIM + 1];
        v4f c100 = lut4[i000 + LUT_DIM2];
        v4f c101 = lut4[i000 + LUT_DIM2 + 1];
        v4f c110 = lut4[i000 + LUT_DIM2 + LUT_DIM];
        v4f c111 = lut4[i000 + LUT_DIM2 + LUT_DIM + 1];

        float w000 = bw0 * gw0 * rw0;
        float w001 = bw0 * gw0 * rw1;
        float w010 = bw0 * gw1 * rw0;
        float w011 = bw0 * gw1 * rw1;
        float w100 = bw1 * gw0 * rw0;
        float w101 = bw1 * gw0 * rw1;
        float w110 = bw1 * gw1 * rw0;
        float w111 = bw1 * gw1 * rw1;

        float orr = w000 * c000.x;
        float og  = w000 * c000.y;
        float ob  = w000 * c000.z;
        orr = fmaf(w001, c001.x, orr); og = fmaf(w001, c001.y, og); ob = fmaf(w001, c001.z, ob);
        orr = fmaf(w010, c010.x, orr); og = fmaf(w010, c010.y, og); ob = fmaf(w010, c010.z, ob);
        orr = fmaf(w011, c011.x, orr); og = fmaf(w011, c011.y, og); ob = fmaf(w011, c011.z, ob);
        orr = fmaf(w100, c100.x, orr); og = fmaf(w100, c100.y, og); ob = fmaf(w100, c100.z, ob);
        orr = fmaf(w101, c101.x, orr); og = fmaf(w101, c101.y, og); ob = fmaf(w101, c101.z, ob);
        orr = fmaf(w110, c110.x, orr); og = fmaf(w110, c110.y, og); ob = fmaf(w110, c110.z, ob);
        orr = fmaf(w111, c111.x, orr); og = fmaf(w111, c111.y, og); ob = fmaf(w111, c111.z, ob);

        outr[j] = orr;
        outg[j] = og;
        outb[j] = ob;
    }

    // Streamed writes: non-temporal (write-once).
    __builtin_nontemporal_store(outr, (v4f*)(out + base));
    __builtin_nontemporal_store(outg, (v4f*)(out + base + HW));
    __builtin_nontemporal_store(outb, (v4f*)(out + base + 2 * HW));
}

// ---- Fallback: gather directly from planar LUT (used only if d_ws too small) ----
__global__ __launch_bounds__(256) void lut_apply_raw_kernel(
    const float* __restrict__ x, const float* __restrict__ lut,
    float* __restrict__ out, int total_quads)
{
    int tid = blockIdx.x * blockDim.x + threadIdx.x;
    if (tid >= total_quads) return;

    int img = tid >> QPI_SHIFT;
    int q   = tid & QPI_MASK;
    size_t base = (size_t)img * 3 * HW + ((size_t)q << 2);

    const v4f r4 = __builtin_nontemporal_load((const v4f*)(x + base));
    const v4f g4 = __builtin_nontemporal_load((const v4f*)(x + base + HW));
    const v4f b4 = __builtin_nontemporal_load((const v4f*)(x + base + 2 * HW));

    v4f outr, outg, outb;
    const float scale = (float)(LUT_DIM - 1);

    #pragma unroll
    for (int j = 0; j < 4; ++j) {
        float rs = r4[j] * scale;
        float gs = g4[j] * scale;
        float bs = b4[j] * scale;

        int r_id = min(max((int)floorf(rs), 0), LUT_DIM - 2);
        int g_id = min(max((int)floorf(gs), 0), LUT_DIM - 2);
        int b_id = min(max((int)floorf(bs), 0), LUT_DIM - 2);

        float rd = rs - (float)r_id;
        float gd = gs - (float)g_id;
        float bd = bs - (float)b_id;

        float orr = 0.0f, og = 0.0f, ob = 0.0f;
        #pragma unroll
        for (int db = 0; db < 2; ++db) {
            float wb = db ? bd : 1.0f - bd;
            #pragma unroll
            for (int dg = 0; dg < 2; ++dg) {
                float wg = dg ? gd : 1.0f - gd;
                #pragma unroll
                for (int dr = 0; dr < 2; ++dr) {
                    float wr = dr ? rd : 1.0f - rd;
                    float w = wb * wg * wr;
                    int idx = ((b_id + db) * LUT_DIM + (g_id + dg)) * LUT_DIM + (r_id + dr);
                    orr = fmaf(w, lut[idx],             orr);
                    og  = fmaf(w, lut[idx + LUT_N],     og);
                    ob  = fmaf(w, lut[idx + 2 * LUT_N], ob);
                }
            }
        }
        outr[j] = orr; outg[j] = og; outb[j] = ob;
    }

    __builtin_nontemporal_store(outr, (v4f*)(out + base));
    __builtin_nontemporal_store(outg, (v4f*)(out + base + HW));
    __builtin_nontemporal_store(outb, (v4f*)(out + base + 2 * HW));
}

extern "C" void kernel_launch(void* const* d_in, const int* in_sizes, int n_in,
                              void* d_out, int out_size, void* d_ws, size_t ws_size,
                              hipStream_t stream) {
    const float* x   = (const float*)d_in[0];   // (32, 3, 512, 512) fp32
    const float* lut = (const float*)d_in[1];   // (3, 33, 33, 33) fp32
    float* out = (float*)d_out;

    long total_px    = (long)in_sizes[0] / 3;   // 8,388,608
    int  total_quads = (int)(total_px >> 2);    // 2,097,152
    int  blocks      = (total_quads + 255) / 256;

    size_t lut4_bytes = (size_t)LUT_N * sizeof(v4f);   // ~575 KB
    if (ws_size >= lut4_bytes) {
        v4f* lut4 = (v4f*)d_ws;
        lut_pack_kernel<<<(LUT_N + 255) / 256, 256, 0, stream>>>(lut, lut4);
        lut_apply_kernel<<<blocks, 256, 0, stream>>>(x, lut4, out, total_quads);
    } else {
        lut_apply_raw_kernel<<<blocks, 256, 0, stream>>>(x, lut, out, total_quads);
    }
}